// DGMGeometryAwareRelationalGraphNeuralNetwork_45775761441170
// MI455X (gfx1250) — compile-verified
//
#include <hip/hip_runtime.h>
#include <hip/hip_bf16.h>

typedef __attribute__((ext_vector_type(2))) float v2f;
typedef __attribute__((ext_vector_type(4))) float v4f;
typedef __attribute__((ext_vector_type(8))) float v8f;

#define N_NODES 30000
#define N_EDGES 600000
#define DIM     128
#define NREL    7
#define NGRAPH  16
#define NLAYER  3
#define KUPD    (NREL * DIM)      // 896
#define KTOT    (KUPD + DIM)      // 1024
#define UPD_ELEMS ((size_t)N_NODES * KUPD)  // 26,880,000 floats

#define MBLK 64
#define NBLK 64
#define KC   64
#define NCHUNK (KTOT / KC)        // 16; chunks 0..13 -> upd/Wr, 14..15 -> h/Wl
#define CSPLIT (KUPD / KC)        // 14

// LDS strides chosen for conflict-free WMMA fragment loads (64 banks x 4B):
//   A row stride 68 floats  -> bank offset 4*row  -> 16 rows x 2 dwords cover
//     32 banks, upper half-wave (K+2) fills the other gaps.
//   B pair-row stride 160 floats -> 160 % 64 == 32 -> upper half-wave's
//     pair-row lands in the opposite 32-bank half.
#define ASTR 68
#define BSTR 160
#define KPAIRS (KC / 2)           // 32 k-pairs per chunk

// ---------------------------------------------------------------- zero fill
__global__ void zero_f4(float* __restrict__ p, int n4) {
    int i = blockIdx.x * blockDim.x + threadIdx.x;
    if (i < n4) {
        v4f z = {};
        *(v4f*)(p + 4 * (size_t)i) = z;
    }
}

// -------------------------------------------------------- edge scatter-add
// one wave (32 lanes) per edge; each lane handles 4 features (float4 gather,
// 4 native global_atomic_add_f32 to the L2-resident upd buffer)
__global__ __launch_bounds__(256) void scatter_edges(
    const float* __restrict__ h,
    const int*   __restrict__ node_in,
    const int*   __restrict__ node_out,
    const int*   __restrict__ relation,
    const float* __restrict__ ew,
    float*       __restrict__ upd)
{
    const int lane = threadIdx.x & 31;
    const int wave = threadIdx.x >> 5;
    const int d4   = lane * 4;
    const int ebase = blockIdx.x * 32 + wave;
#pragma unroll
    for (int it = 0; it < 4; ++it) {
        const int e = ebase + it * 8;          // E = 600000 = 18750*32, no guard
        const int   in  = node_in[e];
        const int   seg = node_out[e] * NREL + relation[e];
        const float w   = ew[e];
        v4f m = *(const v4f*)(h + (size_t)in * DIM + d4);
        float* dst = upd + (size_t)seg * DIM + d4;
        unsafeAtomicAdd(dst + 0, m.x * w);
        unsafeAtomicAdd(dst + 1, m.y * w);
        unsafeAtomicAdd(dst + 2, m.z * w);
        unsafeAtomicAdd(dst + 3, m.w * w);
    }
}

// ------------------------------------------------- fused GEMM + bias + ReLU
// h_new = relu([upd | h] @ [Wr ; Wl] + (br + bl)) using V_WMMA_F32_16X16X4_F32
__global__ __launch_bounds__(256) void gemm_fused(
    const float* __restrict__ upd,   // N x 896
    const float* __restrict__ h,     // N x 128
    const float* __restrict__ Wr,    // 896 x 128
    const float* __restrict__ Wl,    // 128 x 128
    const float* __restrict__ br,    // 128
    const float* __restrict__ bl,    // 128
    float*       __restrict__ out)   // N x 128
{
    __shared__ float As[2][MBLK * ASTR];        // [row*ASTR + k]
    __shared__ float Bs[2][KPAIRS * BSTR];      // [(k/2)*BSTR + n*2 + (k&1)]

    const int tid  = threadIdx.x;
    const int lane = tid & 31;
    const int wave = tid >> 5;
    const int m0 = blockIdx.x * MBLK;
    const int n0 = blockIdx.y * NBLK;

    const int lm   = lane & 15;            // M/N index within fragment
    const int koff = (lane >> 4) << 1;     // lanes 16-31 carry K = k0+2, k0+3
    const int wm   = (wave & 3) * 16;      // wave's row sub-tile in block
    const int wn   = (wave >> 2) * 32;     // wave's col sub-tile in block

    v8f acc0 = {};
    v8f acc1 = {};

    v4f ra[4], rb[4];                      // global->LDS staging registers

    auto gload = [&](int c) {
#pragma unroll
        for (int v = 0; v < 4; ++v) {
            const int lin = (tid + v * 256) * 4;   // 0..4095
            const int rr  = lin >> 6;              // /KC (=/NBLK)
            const int cc  = lin & 63;
            // A: rows of [upd | h]
            const int grow = m0 + rr;
            v4f a = {};
            if (grow < N_NODES) {
                if (c < CSPLIT) a = *(const v4f*)(upd + (size_t)grow * KUPD + c * KC + cc);
                else            a = *(const v4f*)(h   + (size_t)grow * DIM  + (c - CSPLIT) * KC + cc);
            }
            ra[v] = a;
            // B: rows of [Wr ; Wl]
            if (c < CSPLIT) rb[v] = *(const v4f*)(Wr + (size_t)(c * KC + rr) * DIM + n0 + cc);
            else            rb[v] = *(const v4f*)(Wl + (size_t)((c - CSPLIT) * KC + rr) * DIM + n0 + cc);
        }
    };
    auto stage = [&](int buf) {
#pragma unroll
        for (int v = 0; v < 4; ++v) {
            const int lin = (tid + v * 256) * 4;
            const int rr  = lin >> 6;
            const int cc  = lin & 63;
            // A tile: padded row-major, vector store
            *(v4f*)&As[buf][rr * ASTR + cc] = ra[v];
            // B tile: k-pair interleaved so fragments are one ds_load_b64
            const int kp  = rr >> 1;
            const int par = rr & 1;
            float* bb = &Bs[buf][kp * BSTR + par];
            bb[(cc + 0) * 2] = rb[v].x;
            bb[(cc + 1) * 2] = rb[v].y;
            bb[(cc + 2) * 2] = rb[v].z;
            bb[(cc + 3) * 2] = rb[v].w;
        }
    };
    auto compute = [&](int buf) {
        const float* A = As[buf];
        const float* B = Bs[buf];
#pragma unroll
        for (int k0 = 0; k0 < KC; k0 += 4) {
            const int kk = k0 + koff;                  // even
            // A fragment: lane L holds row (L&15), K = kk, kk+1
            v2f a = *(const v2f*)(A + (wm + lm) * ASTR + kk);
            // B fragments: lane L holds col (L&15), K = kk, kk+1 (contiguous)
            const int kp = kk >> 1;
            v2f b0 = *(const v2f*)(B + kp * BSTR + (wn + lm) * 2);
            v2f b1 = *(const v2f*)(B + kp * BSTR + (wn + 16 + lm) * 2);
            acc0 = __builtin_amdgcn_wmma_f32_16x16x4_f32(
                       false, a, false, b0, (short)0, acc0, false, false);
            acc1 = __builtin_amdgcn_wmma_f32_16x16x4_f32(
                       false, a, false, b1, (short)0, acc1, false, false);
        }
    };

    // software pipeline over 16 K-chunks, double-buffered LDS
    gload(0);
    stage(0);
    __syncthreads();
    for (int c = 0; c < NCHUNK; ++c) {
        const int buf = c & 1;
        if (c + 1 < NCHUNK) gload(c + 1);      // issue global loads early
        compute(buf);
        if (c + 1 < NCHUNK) stage(buf ^ 1);
        __syncthreads();
    }

    // epilogue: bias + ReLU + store (C/D layout: row = j + 8*(lane>>4))
    const int ng0 = n0 + wn + lm;
    const int ng1 = ng0 + 16;
    const float bias0 = br[ng0] + bl[ng0];
    const float bias1 = br[ng1] + bl[ng1];
    const int rbase = m0 + wm + ((lane >> 4) << 3);
#pragma unroll
    for (int j = 0; j < 8; ++j) {
        const int row = rbase + j;
        if (row < N_NODES) {
            float v0 = acc0[j] + bias0;
            float v1 = acc1[j] + bias1;
            out[(size_t)row * DIM + ng0] = v0 > 0.f ? v0 : 0.f;
            out[(size_t)row * DIM + ng1] = v1 > 0.f ? v1 : 0.f;
        }
    }
}

// ------------------------------------------------------------ graph pooling
__global__ __launch_bounds__(256) void graph_pool(
    const float* __restrict__ h,
    const int*   __restrict__ n2g,
    float*       __restrict__ gf)
{
    const int lane = threadIdx.x & 31;
    const int wave = threadIdx.x >> 5;
    const int node = blockIdx.x * 8 + wave;    // N = 30000 = 3750*8
    if (node >= N_NODES) return;
    const int g  = n2g[node];
    const int d4 = lane * 4;
    v4f v = *(const v4f*)(h + (size_t)node * DIM + d4);
    float* dst = gf + (size_t)g * DIM + d4;
    unsafeAtomicAdd(dst + 0, v.x);
    unsafeAtomicAdd(dst + 1, v.y);
    unsafeAtomicAdd(dst + 2, v.z);
    unsafeAtomicAdd(dst + 3, v.w);
}

// ---------------------------------------------------------------- launcher
extern "C" void kernel_launch(void* const* d_in, const int* in_sizes, int n_in,
                              void* d_out, int out_size, void* d_ws, size_t ws_size,
                              hipStream_t stream) {
    const float* x        = (const float*)d_in[0];
    const int*   node_in  = (const int*)  d_in[1];
    const int*   node_out = (const int*)  d_in[2];
    const int*   relation = (const int*)  d_in[3];
    const float* ew       = (const float*)d_in[4];
    const int*   n2g      = (const int*)  d_in[5];
    const float* Wr       = (const float*)d_in[6];
    const float* br       = (const float*)d_in[7];
    const float* Wl       = (const float*)d_in[8];
    const float* bl       = (const float*)d_in[9];

    float* gf       = (float*)d_out;              // 16 x 128
    float* nodefeat = gf + NGRAPH * DIM;          // 30000 x 128

    float* updw = (float*)d_ws;                   // 30000 x 896
    float* hbuf = updw + UPD_ELEMS;               // 30000 x 128

    const int updN4 = (int)(UPD_ELEMS / 4);       // 6,720,000
    dim3 gemmGrid((N_NODES + MBLK - 1) / MBLK, DIM / NBLK);  // (469, 2)

    for (int l = 0; l < NLAYER; ++l) {
        const float* hsrc = (l == 0) ? x : hbuf;
        float*       hdst = (l == NLAYER - 1) ? nodefeat : hbuf;
        zero_f4<<<updN4 / 256, 256, 0, stream>>>(updw, updN4);
        scatter_edges<<<N_EDGES / 32, 256, 0, stream>>>(
            hsrc, node_in, node_out, relation, ew, updw);
        gemm_fused<<<gemmGrid, 256, 0, stream>>>(
            updw, hsrc,
            Wr + (size_t)l * KUPD * DIM, Wl + (size_t)l * DIM * DIM,
            br + l * DIM, bl + l * DIM, hdst);
    }

    zero_f4<<<2, 256, 0, stream>>>(gf, (NGRAPH * DIM) / 4);
    graph_pool<<<N_NODES / 8, 256, 0, stream>>>(nodefeat, n2g, gf);
}